// Experts_1099511628053
// MI455X (gfx1250) — compile-verified
//
#include <hip/hip_runtime.h>
#include <hip/hip_bf16.h>
#include <math.h>
#include <stdint.h>

#define DIMD 512
#define NEXP 16
#define SEQL 4096
#define KBIG 1536   // 3*DIM
#define NOUT 8192   // NE*DIM
#define KR   2560   // 5*DIM

#define BLK_M 128
#define BLK_N 64
#define KB 32
#define A_ST 36     // 144B row stride: 16B aligned, bank-conflict-free
#define B_ST 36
#define NTHREADS 256      // 8 waves; each wave owns a 32(M) x 32(N) strip
#define NIT (KBIG / KB)   // 48 K-stages

typedef __attribute__((ext_vector_type(2))) float v2f;
typedef __attribute__((ext_vector_type(8))) float v8f;

// ---------------- Kernel 1: R = concat(h,us,ue) @ W_r^T + b_r  (tiny GEMV)
__global__ void r_kernel(const float* __restrict__ h, const float* __restrict__ us,
                         const float* __restrict__ ue, const float* __restrict__ Wr,
                         const float* __restrict__ br, float* __restrict__ R) {
  int j = blockIdx.x * blockDim.x + threadIdx.x;
  if (j >= DIMD) return;
  const float* w = Wr + (size_t)j * KR;
  float acc = br[j];
#pragma unroll 4
  for (int k = 0; k < 3 * DIMD; ++k) acc = fmaf(h[k], w[k], acc);
#pragma unroll 4
  for (int k = 0; k < DIMD; ++k) acc = fmaf(us[k], w[3 * DIMD + k], acc);
#pragma unroll 4
  for (int k = 0; k < DIMD; ++k) acc = fmaf(ue[k], w[4 * DIMD + k], acc);
  R[j] = acc;
}

// ---------------- CDNA5 async global->LDS copy helpers (ASYNCcnt-tracked)
__device__ __forceinline__ void async_b128(unsigned lds_off, const void* gsrc) {
  unsigned long long g = (unsigned long long)gsrc;
  asm volatile("global_load_async_to_lds_b128 %0, %1, off"
               :: "v"(lds_off), "v"(g) : "memory");
}
__device__ __forceinline__ void wait_async0() {
  asm volatile("s_wait_asynccnt 0x0" ::: "memory");
}

// 16-lane butterfly helpers (xor masks 1,2,4,8 stay within each half-wave group)
__device__ __forceinline__ float bmax16(float x) {
#pragma unroll
  for (int off = 8; off; off >>= 1) x = fmaxf(x, __shfl_xor(x, off, 32));
  return x;
}
__device__ __forceinline__ float bsum16(float x) {
#pragma unroll
  for (int off = 8; off; off >>= 1) x += __shfl_xor(x, off, 32);
  return x;
}

// ---------------- Kernel 2: fused triple GEMM (fp32 WMMA) + expert epilogue
__global__ __launch_bounds__(NTHREADS, 1)
void experts_kernel(const float* __restrict__ u, const float* __restrict__ noise,
                    const float* __restrict__ Wnn, const float* __restrict__ bnn,
                    const float* __restrict__ Wno, const float* __restrict__ bno,
                    const float* __restrict__ WE,  const float* __restrict__ bE,
                    const float* __restrict__ R,   float* __restrict__ out) {
  __shared__ float Als[2][BLK_M * A_ST];
  __shared__ float Bls[2][3][BLK_N * B_ST];

  const int tid   = threadIdx.x;
  const int lane  = tid & 31;
  const int wave  = tid >> 5;      // 0..7
  const int wsub  = wave & 3;      // s sub-strip: 32 rows each
  const int wn    = wave >> 2;     // 0..1 : 32-column o sub
  const int lhalf = lane >> 4;     // 0/1
  const int lm    = lane & 15;

  const int s_blk = blockIdx.y * BLK_M;
  const int o_blk = blockIdx.x * BLK_N;

  // accumulators: [m-subtile(2)][d-subtile(2)][matrix(3)] = 12 x v8f
  v8f acc[2][2][3] = {};
  const float* Wmat[3] = {Wnn, Wno, WE};

  // per-thread staging coordinates (256 threads)
  const int a_row = tid >> 1;            // 0..127
  const int a_kq  = (tid & 1) * 16;      // 0 or 16 (4 x b128 each)
  const int b_col = tid >> 2;            // 0..63
  const int b_kq  = (tid & 3) * 8;       // 0,8,16,24 (2 x b128 per matrix)

  // ---- async stage issue (k0 stage -> buffer buf)
  auto stage_async = [&](int k0, int buf) {
    const float* asrc = (k0 < 2 * DIMD)
        ? (u + (size_t)(s_blk + a_row) * (2 * DIMD) + k0 + a_kq)
        : (R + (k0 - 2 * DIMD) + a_kq);                 // broadcast row
    unsigned adst = (unsigned)(uintptr_t)&Als[buf][a_row * A_ST + a_kq];
#pragma unroll
    for (int q = 0; q < 4; ++q) async_b128(adst + 16 * q, asrc + 4 * q);
#pragma unroll
    for (int m = 0; m < 3; ++m) {
      const float* bsrc = Wmat[m] + (size_t)(o_blk + b_col) * KBIG + k0 + b_kq;
      unsigned bdst = (unsigned)(uintptr_t)&Bls[buf][m][b_col * B_ST + b_kq];
      async_b128(bdst, bsrc);
      async_b128(bdst + 16, bsrc + 4);
    }
  };

  stage_async(0, 0);

  for (int it = 0; it < NIT; ++it) {
    const int cur = it & 1;
    wait_async0();          // batch `it` is the only outstanding async group
    __syncthreads();

    if (it + 1 < NIT) stage_async((it + 1) * KB, cur ^ 1);  // overlaps WMMA below

    // ---- software-pipelined fragment loads: 8 K-steps of 4, 12 WMMA each
    v2f fa0[2], fa1[2], fb[2][2][3];
    auto load_frags = [&](int pb, int kk) {
      const int ab = (wsub * 32 + lm) * A_ST + kk + 2 * lhalf;
      fa0[pb] = *(const v2f*)&Als[cur][ab];
      fa1[pb] = *(const v2f*)&Als[cur][ab + 16 * A_ST];
#pragma unroll
      for (int dsub = 0; dsub < 2; ++dsub) {
        const int cb = (wn * 32 + dsub * 16 + lm) * B_ST + kk + 2 * lhalf;
#pragma unroll
        for (int m = 0; m < 3; ++m)
          fb[pb][dsub][m] = *(const v2f*)&Bls[cur][m][cb];
      }
    };

    load_frags(0, 0);
#pragma unroll
    for (int i = 0; i < KB / 4; ++i) {
      const int pb = i & 1;
      if (i + 1 < KB / 4) load_frags(pb ^ 1, (i + 1) * 4);  // prefetch next step
#pragma unroll
      for (int dsub = 0; dsub < 2; ++dsub) {
#pragma unroll
        for (int m = 0; m < 3; ++m) {
          acc[0][dsub][m] = __builtin_amdgcn_wmma_f32_16x16x4_f32(
              false, fa0[pb], false, fb[pb][dsub][m], (short)0,
              acc[0][dsub][m], false, false);
          acc[1][dsub][m] = __builtin_amdgcn_wmma_f32_16x16x4_f32(
              false, fa1[pb], false, fb[pb][dsub][m], (short)0,
              acc[1][dsub][m], false, false);
        }
      }
    }
    __syncthreads();   // all reads of buffer `cur` done before it is overwritten
  }

  // ---- epilogue: per (s,d) combine 16 experts living in the 16 lanes of a half
#pragma unroll
  for (int msub = 0; msub < 2; ++msub) {
    const int s_row0 = s_blk + wsub * 32 + msub * 16;
#pragma unroll
    for (int dsub = 0; dsub < 2; ++dsub) {
      const int o = o_blk + wn * 32 + dsub * 16 + lm;   // this lane's expert col
      const int d = o >> 4;                              // uniform within half
      const float cb_nn = bnn[o];
      const float cb_no = bno[o];
      const float cb_e  = bE[o];
#pragma unroll
      for (int r = 0; r < 8; ++r) {
        const int s = s_row0 + r + 8 * lhalf;            // C layout: M = r + 8*half
        const float nz = noise[(size_t)s * NOUT + o];    // noise[s,d,n] == s*8192+o
        const float v  = (acc[msub][dsub][0][r] + cb_nn) +
                         (acc[msub][dsub][1][r] + cb_no) * nz;
        const float ev = acc[msub][dsub][2][r] + cb_e;

        // top-2 over the 16 lanes of this half (tie -> lowest index, like top_k)
        const float m1 = bmax16(v);
        unsigned long long bal = __ballot(v == m1);
        unsigned hmask = (unsigned)((bal >> (lhalf * 16)) & 0xffffu);
        const int arg1 = (__ffs(hmask) - 1) + lhalf * 16;
        const float v1 = (lane == arg1) ? -3.402823466e38f : v;
        const float m2 = bmax16(v1);
        unsigned long long bal2 = __ballot(v1 == m2);
        unsigned hmask2 = (unsigned)((bal2 >> (lhalf * 16)) & 0xffffu);
        const int arg2 = (__ffs(hmask2) - 1) + lhalf * 16;

        const bool keep = (lane == arg1) || (lane == arg2);
        const float mv = keep ? v : 0.0f;
        const float t  = (mv == 0.0f) ? (mv - 100000.0f) : mv;  // faithful ==0 fill
        const float tm = bmax16(t);
        const float ex = __expf(t - tm);
        const float ssum = bsum16(ex);
        const float g = ex / ssum;
        const float red = bsum16(g * ev * (1.0f / 16.0f));      // mean over experts
        if (lm == 0) out[(size_t)s * DIMD + d] = red;
      }
    }
  }
}

extern "C" void kernel_launch(void* const* d_in, const int* in_sizes, int n_in,
                              void* d_out, int out_size, void* d_ws, size_t ws_size,
                              hipStream_t stream) {
  const float* h     = (const float*)d_in[0];
  const float* us    = (const float*)d_in[1];
  const float* ue    = (const float*)d_in[2];
  const float* u     = (const float*)d_in[3];
  const float* noise = (const float*)d_in[4];
  const float* Wnn   = (const float*)d_in[5];
  const float* bnn   = (const float*)d_in[6];
  const float* Wno   = (const float*)d_in[7];
  const float* bno   = (const float*)d_in[8];
  const float* WE    = (const float*)d_in[9];
  const float* bE    = (const float*)d_in[10];
  const float* Wr    = (const float*)d_in[11];
  const float* br    = (const float*)d_in[12];
  float* out = (float*)d_out;
  float* R   = (float*)d_ws;   // 512 floats of scratch

  r_kernel<<<2, 256, 0, stream>>>(h, us, ue, Wr, br, R);

  dim3 grid(NOUT / BLK_N, SEQL / BLK_M);
  experts_kernel<<<grid, NTHREADS, 0, stream>>>(u, noise, Wnn, bnn, Wno, bno,
                                                WE, bE, R, out);
}